// GCNWithJK_11055245820055
// MI455X (gfx1250) — compile-verified
//
#include <hip/hip_runtime.h>
#include <hip/hip_bf16.h>

typedef float v2f __attribute__((ext_vector_type(2)));
typedef float v8f __attribute__((ext_vector_type(8)));

#define N_NODES 100000
#define N_EDGES 1600000
#define DH      128
#define N_GRAPHS 64
#define JK      384   // 3*DH

__device__ __forceinline__ void gfadd(float* p, float v) {
    unsafeAtomicAdd(p, v);   // lowers to global_atomic_add_f32 on gfx1250
}

// ---------------------------------------------------------------------------
// deg[dst] += 1 over real edges (self-loop +1 folded into dinv_kernel)
__global__ __launch_bounds__(256) void degree_kernel(const int* __restrict__ dst,
                                                     float* __restrict__ deg, int ne) {
    int e = blockIdx.x * 256 + threadIdx.x;
    if (e < ne) gfadd(&deg[dst[e]], 1.0f);
}

__global__ __launch_bounds__(256) void dinv_kernel(float* __restrict__ dinv, int n) {
    int i = blockIdx.x * 256 + threadIdx.x;
    if (i < n) dinv[i] = rsqrtf(dinv[i] + 1.0f);   // +1 = self loop; always > 0
}

// ---------------------------------------------------------------------------
// hs = (X @ W) * dinv[row], X:[N,128], W:[128,128]. One block = 16-row slab,
// 8 waves, each wave a 16x16 output tile via V_WMMA_F32_16X16X4_F32.
__global__ __launch_bounds__(256) void gemm_scale_kernel(
    const float* __restrict__ X, const float* __restrict__ W,
    const float* __restrict__ dinv, float* __restrict__ hs) {
    __shared__ float sWt[DH * DH];   // 64KB, transposed: sWt[col*128 + k] = W[k][col]

    const int tid  = threadIdx.x;
    const int row0 = blockIdx.x * 16;

    // cooperative transpose-load of W (16384 floats, float4 global reads)
    for (int i = tid; i < DH * (DH / 4); i += 256) {
        float4 w = ((const float4*)W)[i];
        int k = i >> 5;            // row of W
        int c = (i & 31) << 2;     // first col of the 4
        sWt[(c + 0) * DH + k] = w.x;
        sWt[(c + 1) * DH + k] = w.y;
        sWt[(c + 2) * DH + k] = w.z;
        sWt[(c + 3) * DH + k] = w.w;
    }
    __syncthreads();

    const int lane  = tid & 31;
    const int wave  = tid >> 5;
    const int mrow  = lane & 15;    // A row / B col within tile
    const int khalf = lane >> 4;    // 0: K=2k, 1: K=2k+1 pair selector

    const float* aBase = X + (size_t)(row0 + mrow) * DH + 2 * khalf;
    const float* bBase = sWt + (wave * 16 + mrow) * DH + 2 * khalf;

    v8f acc = {};
    #pragma unroll 8
    for (int kk = 0; kk < 32; ++kk) {
        v2f a = *(const v2f*)(aBase + 4 * kk);     // A[m][4kk+2h .. +1]
        v2f b = *(const v2f*)(bBase + 4 * kk);     // W[4kk+2h..][col] (transposed in LDS)
        acc = __builtin_amdgcn_wmma_f32_16x16x4_f32(
            false, a, false, b, (short)0, acc, false, false);
    }

    // D layout: VGPR j -> row j (lanes 0-15) / row 8+j (lanes 16-31), col = lane&15
    const int ncol = wave * 16 + (lane & 15);
    #pragma unroll
    for (int j = 0; j < 8; ++j) {
        int r = row0 + j + 8 * khalf;
        hs[(size_t)r * DH + ncol] = acc[j] * dinv[r];
    }
}

// ---------------------------------------------------------------------------
// acc[dst] += hs[src]; one wave per edge, float4 per lane (32*4 = 128 ch)
__global__ __launch_bounds__(256) void scatter_kernel(
    const int* __restrict__ src, const int* __restrict__ dst,
    const float* __restrict__ hs, float* __restrict__ acc, int ne) {
    int e = blockIdx.x * 8 + (threadIdx.x >> 5);
    if (e >= ne) return;
    int lane = threadIdx.x & 31;
    int s = src[e], d = dst[e];
    float4 v = ((const float4*)(hs + (size_t)s * DH))[lane];
    float* p = acc + (size_t)d * DH + lane * 4;
    gfadd(p + 0, v.x);
    gfadd(p + 1, v.y);
    gfadd(p + 2, v.z);
    gfadd(p + 3, v.w);
}

// ---------------------------------------------------------------------------
// out = relu(dinv*(acc + hs) + b)   (the +hs term is the self-loop message)
__global__ __launch_bounds__(256) void finalize_kernel(
    const float* __restrict__ acc, const float* __restrict__ hs,
    const float* __restrict__ dinv, const float* __restrict__ bias,
    float* __restrict__ out, int n) {
    int idx = blockIdx.x * 256 + threadIdx.x;   // over n*32 float4
    if (idx >= n * 32) return;
    int node = idx >> 5, c4 = idx & 31;
    float di  = dinv[node];
    float4 a  = ((const float4*)acc)[idx];
    float4 h  = ((const float4*)hs)[idx];
    float4 b  = ((const float4*)bias)[c4];
    float4 o;
    o.x = fmaxf(fmaf(di, a.x + h.x, b.x), 0.0f);
    o.y = fmaxf(fmaf(di, a.y + h.y, b.y), 0.0f);
    o.z = fmaxf(fmaf(di, a.z + h.z, b.z), 0.0f);
    o.w = fmaxf(fmaf(di, a.w + h.w, b.w), 0.0f);
    ((float4*)out)[idx] = o;
}

// ---------------------------------------------------------------------------
// segment-sum into pooled[g][off+c]; batch is SORTED, so keep a running sum
// and flush via atomic only on graph change. 128 threads = 128 channels.
__global__ __launch_bounds__(128) void pool_kernel(
    const float* __restrict__ h, const int* __restrict__ batch,
    float* __restrict__ pooled, int n, int off, int chunk) {
    int start = blockIdx.x * chunk;
    int end   = min(start + chunk, n);
    if (start >= end) return;
    int c = threadIdx.x;
    float run = 0.0f;
    int gprev = batch[start];
    for (int i = start; i < end; ++i) {
        int g = batch[i];
        if (g != gprev) {
            gfadd(&pooled[gprev * JK + off + c], run);
            run = 0.0f;
            gprev = g;
        }
        run += h[(size_t)i * DH + c];
    }
    gfadd(&pooled[gprev * JK + off + c], run);
}

__global__ __launch_bounds__(256) void counts_kernel(
    const int* __restrict__ batch, float* __restrict__ counts, int n) {
    __shared__ float sc[N_GRAPHS];
    if (threadIdx.x < N_GRAPHS) sc[threadIdx.x] = 0.0f;
    __syncthreads();
    for (int i = threadIdx.x; i < n; i += 256) atomicAdd(&sc[batch[i]], 1.0f);
    __syncthreads();
    if (threadIdx.x < N_GRAPHS) counts[threadIdx.x] = sc[threadIdx.x];
}

// ---------------------------------------------------------------------------
// per-graph MLP head: mean -> lin1+relu -> lin2 -> log_softmax. 1 block/graph.
__global__ __launch_bounds__(128) void head_kernel(
    const float* __restrict__ pooled, const float* __restrict__ counts,
    const float* __restrict__ w1, const float* __restrict__ b1,
    const float* __restrict__ w2, const float* __restrict__ b2,
    float* __restrict__ out) {
    __shared__ float sp[JK];
    __shared__ float sg[DH];
    __shared__ float sl[10];
    __shared__ float sred[2];
    int row = blockIdx.x, c = threadIdx.x;
    float inv = 1.0f / fmaxf(counts[row], 1.0f);
    for (int i = c; i < JK; i += 128) sp[i] = pooled[row * JK + i] * inv;
    __syncthreads();
    float a = b1[c];
    for (int k = 0; k < JK; ++k) a = fmaf(sp[k], w1[k * DH + c], a);
    sg[c] = fmaxf(a, 0.0f);
    __syncthreads();
    if (c < 10) {
        float a2 = b2[c];
        for (int k = 0; k < DH; ++k) a2 = fmaf(sg[k], w2[k * 10 + c], a2);
        sl[c] = a2;
    }
    __syncthreads();
    if (c == 0) {
        float m = sl[0];
        for (int i = 1; i < 10; ++i) m = fmaxf(m, sl[i]);
        float s = 0.0f;
        for (int i = 0; i < 10; ++i) s += __expf(sl[i] - m);
        sred[0] = m; sred[1] = logf(s);
    }
    __syncthreads();
    if (c < 10) out[row * 10 + c] = sl[c] - sred[0] - sred[1];
}

// ---------------------------------------------------------------------------
extern "C" void kernel_launch(void* const* d_in, const int* in_sizes, int n_in,
                              void* d_out, int out_size, void* d_ws, size_t ws_size,
                              hipStream_t stream) {
    const float* x     = (const float*)d_in[0];
    const int*   ei    = (const int*)d_in[1];
    const int*   batch = (const int*)d_in[2];
    const float* W1 = (const float*)d_in[4];  const float* b1 = (const float*)d_in[5];
    const float* W2 = (const float*)d_in[6];  const float* b2 = (const float*)d_in[7];
    const float* W3 = (const float*)d_in[8];  const float* b3 = (const float*)d_in[9];
    const float* l1w = (const float*)d_in[10]; const float* l1b = (const float*)d_in[11];
    const float* l2w = (const float*)d_in[12]; const float* l2b = (const float*)d_in[13];
    float* out = (float*)d_out;

    const int N = N_NODES, E = N_EDGES;
    const int* src = ei;
    const int* dst = ei + E;

    char* ws = (char*)d_ws;
    auto alloc = [&](size_t bytes) {
        char* p = ws;
        ws += (bytes + 255) & ~(size_t)255;
        return p;
    };
    const size_t FEAT = (size_t)N * DH * sizeof(float);
    float* dinv   = (float*)alloc((size_t)N * sizeof(float));
    float* hs     = (float*)alloc(FEAT);
    float* acc    = (float*)alloc(FEAT);
    float* h1     = (float*)alloc(FEAT);
    float* h2     = (float*)alloc(FEAT);
    float* h3     = (float*)alloc(FEAT);
    float* pooled = (float*)alloc((size_t)N_GRAPHS * JK * sizeof(float));
    float* counts = (float*)alloc((size_t)N_GRAPHS * sizeof(float));

    // degree -> dinv
    hipMemsetAsync(dinv, 0, (size_t)N * sizeof(float), stream);
    degree_kernel<<<(E + 255) / 256, 256, 0, stream>>>(dst, dinv, E);
    dinv_kernel<<<(N + 255) / 256, 256, 0, stream>>>(dinv, N);

    const float* act = x;
    const float* Ws[3] = {W1, W2, W3};
    const float* bs[3] = {b1, b2, b3};
    float* hout[3] = {h1, h2, h3};
    for (int l = 0; l < 3; ++l) {
        gemm_scale_kernel<<<N / 16, 256, 0, stream>>>(act, Ws[l], dinv, hs);
        hipMemsetAsync(acc, 0, FEAT, stream);
        scatter_kernel<<<(E + 7) / 8, 256, 0, stream>>>(src, dst, hs, acc, E);
        finalize_kernel<<<(N * 32 + 255) / 256, 256, 0, stream>>>(acc, hs, dinv,
                                                                  bs[l], hout[l], N);
        act = hout[l];
    }

    hipMemsetAsync(pooled, 0, (size_t)N_GRAPHS * JK * sizeof(float), stream);
    const int CHUNK = 256;
    int pblocks = (N + CHUNK - 1) / CHUNK;
    pool_kernel<<<pblocks, 128, 0, stream>>>(h1, batch, pooled, N, 0,   CHUNK);
    pool_kernel<<<pblocks, 128, 0, stream>>>(h2, batch, pooled, N, 128, CHUNK);
    pool_kernel<<<pblocks, 128, 0, stream>>>(h3, batch, pooled, N, 256, CHUNK);
    counts_kernel<<<1, 256, 0, stream>>>(batch, counts, N);

    head_kernel<<<N_GRAPHS, 128, 0, stream>>>(pooled, counts, l1w, l1b, l2w, l2b, out);
}